// FCOS_59442347376764
// MI455X (gfx1250) — compile-verified
//
#include <hip/hip_runtime.h>
#include <math.h>

// ---------------- problem constants (match reference) ----------------
#define NUM_B      16
#define NUM_C      80
#define NTOT       17064          // 12800+3200+800+208+56
#define BINS       8192           // bin = float_bits >> 17 (<= 8128 for s<=1.0)
#define CAP        2048           // per-image candidate cap (>= 1000 + hist[T])
#define TOPK       1000
#define MAXDET     100
#define NMS_THR    0.6f
#define PRE_THR    0.05f

__constant__ int c_off[6] = {0, 12800, 16000, 16800, 17008, 17064};
__constant__ int c_H[5]   = {100, 50, 25, 13, 7};
__constant__ int c_W[5]   = {128, 64, 32, 16, 8};
__constant__ int c_S[5]   = {8, 16, 32, 64, 128};

struct Ptrs {
    const float* lg[5];
    const float* bb[5];
    const float* ct[5];
    const float* im;
};

__device__ __forceinline__ float sigmoidf_(float x) {
    return 1.0f / (1.0f + expf(-x));
}

__device__ __forceinline__ int find_level(int g) {
    int l = 0;
    while (l < 4 && g >= c_off[l + 1]) ++l;
    return l;
}

// ---------------- kernel 0: zero histogram ----------------
__global__ void k_zero(int* p, int n) {
    for (int i = blockIdx.x * blockDim.x + threadIdx.x; i < n;
         i += gridDim.x * blockDim.x)
        p[i] = 0;
}

// ---------------- kernel 1: per-image score histogram ----------------
// One thread per (image, location); loops over 80 classes (coalesced per class plane).
__global__ void k_hist(Ptrs P, int* __restrict__ hist) {
    int g = blockIdx.x * blockDim.x + threadIdx.x;
    int b = blockIdx.y;
    if (g >= NTOT) return;
    int l  = find_level(g);
    int p  = g - c_off[l];
    int HW = c_H[l] * c_W[l];
    float cen = sigmoidf_(P.ct[l][b * HW + p]);
    const float* lg = P.lg[l] + (size_t)b * NUM_C * HW + p;
    int* h = hist + b * BINS;
    for (int c = 0; c < NUM_C; ++c) {
        if (c + 1 < NUM_C)                       // gfx1250 global_prefetch_b8
            __builtin_prefetch(lg + (size_t)(c + 1) * HW, 0, 1);
        float cls = sigmoidf_(lg[(size_t)c * HW]);
        if (cls > PRE_THR) {
            float s = cls * cen;                 // s in (0,1)
            unsigned bin = __float_as_uint(s) >> 17;
            if (bin >= BINS) bin = BINS - 1;
            atomicAdd(&h[bin], 1);
        }
    }
}

// ---------------- kernel 2: find threshold bin (suffix count >= TOPK) ----------------
__global__ void k_thresh(const int* __restrict__ hist, int* threshBin, int* candCount) {
    int b = blockIdx.x;
    if (threadIdx.x == 0) {
        const int* h = hist + b * BINS;
        int acc = 0, T = 0;
        for (int i = BINS - 1; i >= 0; --i) {
            acc += h[i];
            if (acc >= TOPK) { T = i; break; }
        }
        threshBin[b] = T;
        candCount[b] = 0;   // reset append counter for this call
    }
}

// ---------------- kernel 3: compact candidates above threshold ----------------
__global__ void k_compact(Ptrs P, const int* __restrict__ threshBin,
                          int* __restrict__ candCount,
                          float* __restrict__ candScore, int* __restrict__ candIdx) {
    int g = blockIdx.x * blockDim.x + threadIdx.x;
    int b = blockIdx.y;
    if (g >= NTOT) return;
    int T  = threshBin[b];
    int l  = find_level(g);
    int p  = g - c_off[l];
    int HW = c_H[l] * c_W[l];
    float cen = sigmoidf_(P.ct[l][b * HW + p]);
    const float* lg = P.lg[l] + (size_t)b * NUM_C * HW + p;
    for (int c = 0; c < NUM_C; ++c) {
        float cls = sigmoidf_(lg[(size_t)c * HW]);
        if (cls > PRE_THR) {
            float s = cls * cen;
            int bin = (int)(__float_as_uint(s) >> 17);
            if (bin >= BINS) bin = BINS - 1;
            if (bin >= T) {
                int pos = atomicAdd(&candCount[b], 1);
                if (pos < CAP) {
                    candScore[b * CAP + pos] = s;
                    candIdx[b * CAP + pos]   = g * NUM_C + c;  // loc*C + cls (top_k flat index)
                }
            }
        }
    }
}

// ---------------- TDM helper: 1-D global->LDS tile load (gfx1250 Tensor Data Mover) ---
#if defined(__gfx1250__) && __has_builtin(__builtin_amdgcn_tensor_load_to_lds)
#define HAVE_TDM 1
typedef unsigned int tdm_v4u __attribute__((ext_vector_type(4)));
typedef int          tdm_v8i __attribute__((ext_vector_type(8)));
typedef int          tdm_v4i __attribute__((ext_vector_type(4)));

__device__ __forceinline__ void tdm_load_1d(unsigned lds_byte,
                                            unsigned long long gaddr,
                                            unsigned nelem /* 4B elems, <= 65535 */) {
    // D# group 0: count=1 | lds_addr | global_addr[56:0] | type=2
    tdm_v4u g0;
    g0.x = 1u;
    g0.y = lds_byte;
    g0.z = (unsigned)(gaddr & 0xFFFFFFFFull);
    g0.w = (unsigned)((gaddr >> 32) & 0x1FFFFFFull) | (2u << 30);
    // D# group 1: data_size=4B, tensor_dim0=nelem, tensor_dim1=1,
    //             tile_dim0=nelem, tile_dim1/2=0, dim0_stride=nelem
    tdm_v8i g1;
    g1[0] = (int)(2u << 16);                         // data_size=2 (4 bytes)
    g1[1] = (int)((nelem & 0xFFFFu) << 16);          // tensor_dim0[15:0] @ bit48
    g1[2] = (int)((nelem >> 16) | (1u << 16));       // tensor_dim0 hi | tensor_dim1 lo (=1)
    g1[3] = (int)((nelem & 0xFFFFu) << 16);          // tensor_dim1 hi (=0) | tile_dim0
    g1[4] = 0;                                       // tile_dim1=0, tile_dim2=0
    g1[5] = (int)nelem;                              // tensor_dim0_stride lo32
    g1[6] = 0;
    g1[7] = 0;
    tdm_v4i z = {0, 0, 0, 0};
#if defined(__clang_major__) && (__clang_major__ >= 23)
    tdm_v8i z8 = {0, 0, 0, 0, 0, 0, 0, 0};
    __builtin_amdgcn_tensor_load_to_lds(g0, g1, z, z, z8, 0);
#else
    __builtin_amdgcn_tensor_load_to_lds(g0, g1, z, z, 0);
#endif
}
#else
#define HAVE_TDM 0
#endif

// ---------------- kernel 4: per-image sort + box decode + sequential NMS ----------------
__global__ __launch_bounds__(1024) void k_sortnms(Ptrs P,
                                                  const float* __restrict__ candScore,
                                                  const int* __restrict__ candIdx,
                                                  const int* __restrict__ candCount,
                                                  float* __restrict__ out) {
    __shared__ float skey[CAP];       // sort keys; later argmax scratch
    __shared__ int   sidx[CAP];
    __shared__ float bx0[1024], bx1[1024], bx2[1024], bx3[1024];
    __shared__ float offv[1024], area[1024], scAct[1024];
    __shared__ int   clsi[1024];
    __shared__ float selS[MAXDET];
    __shared__ int   selI[MAXDET];

    const int b   = blockIdx.x;
    const int tid = threadIdx.x;
    int n = candCount[b];
    if (n > CAP) n = CAP;

#if HAVE_TDM
    // Async stage candidate arrays global -> LDS via Tensor Data Mover (wave 0 only),
    // completion tracked with TENSORcnt.
    if (tid < 32) {
        tdm_load_1d((unsigned)(unsigned long long)(void*)&skey[0],
                    (unsigned long long)(const void*)(candScore + (size_t)b * CAP), CAP);
        tdm_load_1d((unsigned)(unsigned long long)(void*)&sidx[0],
                    (unsigned long long)(const void*)(candIdx + (size_t)b * CAP), CAP);
#if __has_builtin(__builtin_amdgcn_s_wait_tensorcnt)
        __builtin_amdgcn_s_wait_tensorcnt(0);
#endif
    }
    __syncthreads();
#endif
    // Deterministic fill (idempotent w.r.t. the TDM copy above).
    for (int i = tid; i < CAP; i += 1024) {
        if (i < n) {
            skey[i] = candScore[b * CAP + i];
            sidx[i] = candIdx[b * CAP + i];
        } else {
            skey[i] = -INFINITY;
            sidx[i] = 0x7FFFFFFF;
        }
    }
    __syncthreads();

    // Bitonic sort, descending by (score, then ascending index) -> matches lax.top_k ties.
    for (int k = 2; k <= CAP; k <<= 1) {
        for (int j = k >> 1; j > 0; j >>= 1) {
            for (int i = tid; i < CAP; i += 1024) {
                int pj = i ^ j;
                if (pj > i) {
                    float sa = skey[i], sb = skey[pj];
                    int   ia = sidx[i], ib = sidx[pj];
                    bool aFirst = (sa > sb) || (sa == sb && ia < ib);
                    bool doSwap = ((i & k) == 0) ? (!aFirst) : aFirst;
                    if (doSwap) {
                        skey[i] = sb; skey[pj] = sa;
                        sidx[i] = ib; sidx[pj] = ia;
                    }
                }
            }
            __syncthreads();
        }
    }

    // Decode top-1000 boxes, clip, sqrt scores, class offsets, areas.
    {
        float x1 = 0.f, y1 = 0.f, x2 = 0.f, y2 = 0.f;
        float sc = 0.f, off = 0.f, ar = 0.f;
        int cc = 0;
        if (tid < TOPK && tid < n) {
            int idx = sidx[tid];
            int loc = idx / NUM_C;
            cc = idx - loc * NUM_C;
            int l  = find_level(loc);
            int p  = loc - c_off[l];
            int Wl = c_W[l], st = c_S[l];
            int HW = c_H[l] * Wl;
            int yy = p / Wl, xx = p - yy * Wl;
            float lx = (float)(xx * st + (st >> 1));
            float ly = (float)(yy * st + (st >> 1));
            const float* bb = P.bb[l] + (size_t)b * 4 * HW + p;
            float r0 = bb[0], r1 = bb[HW], r2 = bb[2 * HW], r3 = bb[3 * HW];
            float hI = P.im[b * 2 + 0], wI = P.im[b * 2 + 1];
            x1 = fminf(fmaxf(lx - r0, 0.f), wI - 1.f);
            y1 = fminf(fmaxf(ly - r1, 0.f), hI - 1.f);
            x2 = fminf(fmaxf(lx + r2, 0.f), wI - 1.f);
            y2 = fminf(fmaxf(ly + r3, 0.f), hI - 1.f);
            float v = skey[tid];
            sc = (v > 0.f) ? sqrtf(fmaxf(v, 0.f) + 1e-12f) : 0.f;
            off = (float)cc * 10000.f;
            float nx1 = x1 + off, ny1 = y1 + off, nx2 = x2 + off, ny2 = y2 + off;
            ar = fmaxf(nx2 - nx1, 0.f) * fmaxf(ny2 - ny1, 0.f);
        }
        bx0[tid] = x1; bx1[tid] = y1; bx2[tid] = x2; bx3[tid] = y2;
        offv[tid] = off; area[tid] = ar; scAct[tid] = sc; clsi[tid] = cc;
    }
    __syncthreads();

    // Sequential NMS: 100 iterations of (argmax over 1000, suppress IoU >= 0.6).
    for (int it = 0; it < MAXDET; ++it) {
        skey[tid] = (tid < TOPK) ? scAct[tid] : -INFINITY;
        sidx[tid] = tid;
        __syncthreads();
        for (int strd = 512; strd >= 1; strd >>= 1) {
            if (tid < strd) {
                float s1 = skey[tid], s2 = skey[tid + strd];
                int   i1 = sidx[tid], i2 = sidx[tid + strd];
                if (s2 > s1 || (s2 == s1 && i2 < i1)) { skey[tid] = s2; sidx[tid] = i2; }
            }
            __syncthreads();
        }
        int   bi = sidx[0];
        float bs = skey[0];
        if (tid == 0) { selI[it] = bi; selS[it] = bs; }
        float ox  = offv[bi];
        float aI  = area[bi];
        float X1  = bx0[bi] + ox, Y1 = bx1[bi] + ox;
        float X2  = bx2[bi] + ox, Y2 = bx3[bi] + ox;
        if (tid < TOPK) {
            float o2  = offv[tid];
            float xx1 = fmaxf(X1, bx0[tid] + o2);
            float yy1 = fmaxf(Y1, bx1[tid] + o2);
            float xx2 = fminf(X2, bx2[tid] + o2);
            float yy2 = fminf(Y2, bx3[tid] + o2);
            float inter = fmaxf(xx2 - xx1, 0.f) * fmaxf(yy2 - yy1, 0.f);
            float iou   = inter / (aI + area[tid] - inter + 1e-9f);
            if (iou >= NMS_THR || tid == bi) scAct[tid] = -INFINITY;
        }
        __syncthreads();
    }

    // Emit detections: (x1,y1,x2,y2,score,class) or zeros if invalid.
    for (int t = tid; t < MAXDET; t += 1024) {
        int   o = (b * MAXDET + t) * 6;
        int   i = selI[t];
        float s = selS[t];
        bool  valid = s > 0.f;
        out[o + 0] = valid ? bx0[i] : 0.f;
        out[o + 1] = valid ? bx1[i] : 0.f;
        out[o + 2] = valid ? bx2[i] : 0.f;
        out[o + 3] = valid ? bx3[i] : 0.f;
        out[o + 4] = valid ? s : 0.f;
        out[o + 5] = valid ? (float)clsi[i] : 0.f;
    }
}

// ---------------- launch ----------------
extern "C" void kernel_launch(void* const* d_in, const int* in_sizes, int n_in,
                              void* d_out, int out_size, void* d_ws, size_t ws_size,
                              hipStream_t stream) {
    (void)in_sizes; (void)n_in; (void)out_size; (void)ws_size;
    Ptrs P;
    for (int l = 0; l < 5; ++l) {
        P.lg[l] = (const float*)d_in[3 * l + 0];
        P.bb[l] = (const float*)d_in[3 * l + 1];
        P.ct[l] = (const float*)d_in[3 * l + 2];
    }
    P.im = (const float*)d_in[15];

    // Workspace layout (~786 KB total).
    char*  ws        = (char*)d_ws;
    int*   hist      = (int*)ws;                          // 16*8192 ints  = 512 KB
    int*   threshBin = (int*)(ws + (size_t)NUM_B * BINS * 4);
    int*   candCount = threshBin + NUM_B;
    float* candScore = (float*)(candCount + NUM_B);       // 16*2048 floats = 128 KB
    int*   candIdx   = (int*)(candScore + (size_t)NUM_B * CAP); // 128 KB

    const int locBlocks = (NTOT + 255) / 256;  // 67

    k_zero<<<512, 256, 0, stream>>>(hist, NUM_B * BINS);
    k_hist<<<dim3(locBlocks, NUM_B), 256, 0, stream>>>(P, hist);
    k_thresh<<<NUM_B, 32, 0, stream>>>(hist, threshBin, candCount);
    k_compact<<<dim3(locBlocks, NUM_B), 256, 0, stream>>>(P, threshBin, candCount,
                                                          candScore, candIdx);
    k_sortnms<<<NUM_B, 1024, 0, stream>>>(P, candScore, candIdx, candCount,
                                          (float*)d_out);
}